// LinearSelfAttention_2808908611864
// MI455X (gfx1250) — compile-verified
//
#include <hip/hip_runtime.h>

// ---------------------------------------------------------------------------
// LinearSelfAttention, reassociated (5000x fewer FLOPs than naive HtQH):
//   attn = (P H) M H^T Q H,  M = diag(1..1,0)  -> drop column s = n
//   1) PHm = P @ H[:, :n]        (257 x 2048)
//   2) A   = PHm @ H[:, :n]^T    (257 x 257)
//   3) C   = A @ Q               (257 x 257)
//   4) out = H + (C @ H) / n     (257 x 2049), fused epilogue
// ~1.35e10 f32 FLOPs; H (33.7MB) + intermediates fit in 192MB L2.
// f32 kept end-to-end via V_WMMA_F32_16X16X4_F32 (wave32).
//
// Block = 128 thr (4 waves) -> 64x64 output tile, K chunked by 32 through LDS.
// Staging: GLOBAL_LOAD_ASYNC_TO_LDS_B32 (ASYNCcnt) for full-K chunks,
//          sync zero-filling path only for the K tail chunk.
// Hot loop per k-step: ds_load_2addr_b64 + 4x v_wmma, no branches/selects.
// Edge policy: clamp row/col loads (garbage lands only in unstored rows/cols),
// zero-fill k>=K at staging time (the only case that corrupts valid outputs).
// ---------------------------------------------------------------------------

typedef float v2f __attribute__((ext_vector_type(2)));
typedef float v8f __attribute__((ext_vector_type(8)));

#define DP1  257     // d+1
#define NP1  2049    // n+1
#define NCOL 2048    // n
#define NB   16      // batch

#define KC   32                  // K chunk staged in LDS
#define LSTR 34                  // KC + 2 pad: bank-conflict-free b64 reads

#if defined(__has_builtin)
#  if __has_builtin(__builtin_amdgcn_global_load_async_to_lds_b32)
#    define ASYNC_LDS 1
#  endif
#  if __has_builtin(__builtin_amdgcn_s_wait_asynccnt)
#    define ASYNC_WAIT_BUILTIN 1
#  endif
#endif

// Builtin signature (from hipcc diagnostic): param0 = int addrspace(1)*,
// param1 = int addrspace(3)*, then imm offset + imm cpol.
typedef __attribute__((address_space(1))) int* as1_iptr;
typedef __attribute__((address_space(3))) int* as3_iptr;

// Copy 4 bytes global -> LDS. Async (ASYNCcnt-tracked, bypasses VGPRs) when
// the gfx1250 builtin is available; plain copy otherwise.
__device__ __forceinline__ void stage_b32(const float* g, float* l) {
#if ASYNC_LDS
    __builtin_amdgcn_global_load_async_to_lds_b32(
        (as1_iptr)(int*)(float*)g, (as3_iptr)(int*)l, 0, 0);
#else
    *l = *g;
#endif
}

__device__ __forceinline__ void stage_wait() {
#if ASYNC_LDS
#  if ASYNC_WAIT_BUILTIN
    __builtin_amdgcn_s_wait_asynccnt(0);
#  else
    asm volatile("s_wait_asynccnt 0x0" ::: "memory");
#  endif
#endif
}

// Fragment layouts (ISA 7.12.2, wave32), f32 16x16x4:
//   A: lane<16 -> row=lane, {K=k,k+1}; lane>=16 -> row=lane-16, {K=k+2,k+3}
//   B: same with row->col
//   C/D: vgpr i -> row = i + 8*(lane>=16), col = lane&15
template <bool TRANS_B, bool EPILOGUE_H>
__global__ __launch_bounds__(128) void wmma_gemm_f32(
    const float* __restrict__ A, long long sAb, int lda,
    const float* __restrict__ B, long long sBb, int ldb,
    float* __restrict__ D, long long sDb, int ldd,
    const float* __restrict__ Hx, long long sHb, int ldh,
    int M, int N, int K, float alpha)
{
    __shared__ __align__(16) float As[64 * LSTR];  // As[row_local][k_local]
    __shared__ __align__(16) float Bs[64 * LSTR];  // Bs[col_local][k_local]

    const int tid  = threadIdx.x;
    const int wave = tid >> 5;
    const int lane = tid & 31;
    const int half = lane >> 4;
    const int r    = lane & 15;

    const int row0 = blockIdx.x * 64;
    const int col0 = blockIdx.y * 64;
    const int b    = blockIdx.z;

    const float* __restrict__ Ab = A + (long long)b * sAb;
    const float* __restrict__ Bb = B + (long long)b * sBb;
    float*       __restrict__ Db = D + (long long)b * sDb;

    v8f acc[4];
    #pragma unroll
    for (int t = 0; t < 4; ++t) acc[t] = (v8f){0, 0, 0, 0, 0, 0, 0, 0};

    // Staging coordinates (coalesced global reads, conflict-free LDS writes):
    const int a_row = tid >> 5;        // A: 4 rows/pass, k = lane (contig)
    const int a_k   = tid & 31;
    const int bt_c  = tid >> 5;        // B^T: 4 cols/pass, k = lane (contig)
    const int bt_k  = tid & 31;
    const int bn_k  = tid >> 6;        // B:   2 k-rows/pass, col contig
    const int bn_c  = tid & 63;

    // Row/col clamped source offsets are loop-invariant.
    const long long aRowOff  = (long long)min(row0 + a_row, M - 1) * lda;
    const long long btColOff = (long long)min(col0 + bt_c, N - 1) * ldb;
    const int bn_cc = min(col0 + bn_c, N - 1);

    for (int kbase = 0; kbase < K; kbase += KC) {
        const bool kfull = (kbase + KC <= K);
        if (kfull) {
            // ---- async staging: no k masking needed ----
            #pragma unroll
            for (int j = 0; j < 16; ++j) {
                const int rowl = a_row + j * 4;
                stage_b32(&Ab[(long long)min(row0 + rowl, M - 1) * lda + kbase + a_k],
                          &As[rowl * LSTR + a_k]);
            }
            if (TRANS_B) {
                #pragma unroll
                for (int j = 0; j < 16; ++j) {
                    const int coll = bt_c + j * 4;
                    stage_b32(&Bb[(long long)min(col0 + coll, N - 1) * ldb + kbase + bt_k],
                              &Bs[coll * LSTR + bt_k]);
                }
            } else {
                #pragma unroll
                for (int j = 0; j < 16; ++j) {
                    const int kl = bn_k + j * 2;
                    stage_b32(&Bb[(long long)(kbase + kl) * ldb + bn_cc],
                              &Bs[bn_c * LSTR + kl]);
                }
            }
            stage_wait();
        } else {
            // ---- K tail: sync staging with zero fill for k >= K ----
            #pragma unroll
            for (int j = 0; j < 16; ++j) {
                const int rowl = a_row + j * 4;
                const int kg   = kbase + a_k;
                float v = Ab[(long long)min(row0 + rowl, M - 1) * lda + min(kg, K - 1)];
                if (kg >= K) v = 0.0f;
                As[rowl * LSTR + a_k] = v;
            }
            if (TRANS_B) {
                #pragma unroll
                for (int j = 0; j < 16; ++j) {
                    const int coll = bt_c + j * 4;
                    const int kg   = kbase + bt_k;
                    float v = Bb[(long long)min(col0 + coll, N - 1) * ldb + min(kg, K - 1)];
                    if (kg >= K) v = 0.0f;
                    Bs[coll * LSTR + bt_k] = v;
                }
            } else {
                #pragma unroll
                for (int j = 0; j < 16; ++j) {
                    const int kl = bn_k + j * 2;
                    const int kg = kbase + kl;
                    float v = Bb[(long long)min(kg, K - 1) * ldb + bn_cc];
                    if (kg >= K) v = 0.0f;
                    Bs[bn_c * LSTR + kl] = v;
                }
            }
        }
        __syncthreads();

        // Prefetch next chunk toward L0/L2 while we compute this one.
        if (kbase + KC < K) {
            __builtin_prefetch(&Ab[aRowOff + min(kbase + KC + a_k, K - 1)], 0, 1);
            __builtin_prefetch(TRANS_B
                ? &Bb[btColOff + min(kbase + KC + bt_k, K - 1)]
                : &Bb[(long long)min(kbase + KC + bn_k, K - 1) * ldb + bn_cc],
                0, 1);
        }

        // ---- compute: 8 k-steps, each ds_load_2addr_b64 feeds 4x wmma ----
        const float* __restrict__ Arow = &As[(wave * 16 + r) * LSTR];
        #pragma unroll
        for (int kk = 0; kk < KC; kk += 4) {
            const int ka = kk + 2 * half;                 // even -> 8B aligned
            const v2f a = *reinterpret_cast<const v2f*>(Arow + ka);
            #pragma unroll
            for (int t = 0; t < 4; ++t) {
                const v2f bf = *reinterpret_cast<const v2f*>(
                    &Bs[(t * 16 + r) * LSTR + ka]);
                acc[t] = __builtin_amdgcn_wmma_f32_16x16x4_f32(
                    false, a, false, bf, (short)0, acc[t], false, false);
            }
        }
        __syncthreads();
    }

    // ---- epilogue / store (only in-range elements) ----
    const int roww = row0 + wave * 16;
    #pragma unroll
    for (int t = 0; t < 4; ++t) {
        const int col = col0 + t * 16 + r;
        if (col >= N) continue;
        #pragma unroll
        for (int i = 0; i < 8; ++i) {
            const int row = roww + i + 8 * half;
            if (row < M) {
                float v = acc[t][i] * alpha;
                if (EPILOGUE_H)
                    v += Hx[(long long)b * sHb + (long long)row * ldh + col];
                Db[(long long)row * ldd + col] = v;
            }
        }
    }
}

extern "C" void kernel_launch(void* const* d_in, const int* in_sizes, int n_in,
                              void* d_out, int out_size, void* d_ws, size_t ws_size,
                              hipStream_t stream)
{
    const float* H = (const float*)d_in[0];   // (16, 257, 2049)
    const float* P = (const float*)d_in[1];   // (257, 257)
    const float* Q = (const float*)d_in[2];   // (257, 257)
    float*     out = (float*)d_out;           // (16, 257, 2049)

    // Workspace layout (all f32, ~42.1 MB total)
    float* PHm  = (float*)d_ws;                       // 16 x 257 x 2048
    float* Abuf = PHm  + (size_t)NB * DP1 * NCOL;     // 16 x 257 x 257
    float* Cbuf = Abuf + (size_t)NB * DP1 * DP1;      // 16 x 257 x 257

    const long long sH  = (long long)DP1 * NP1;
    const long long sPH = (long long)DP1 * NCOL;
    const long long sAA = (long long)DP1 * DP1;
    const dim3 blk(128);

    // 1) PHm = P @ H[:, :2048]          M=257 N=2048 K=257
    wmma_gemm_f32<false, false><<<dim3(5, 32, NB), blk, 0, stream>>>(
        P, 0, DP1, H, sH, NP1, PHm, sPH, NCOL,
        nullptr, 0, 0, DP1, NCOL, DP1, 1.0f);

    // 2) A = PHm @ H[:, :2048]^T        M=257 N=257 K=2048  (mask via K range)
    wmma_gemm_f32<true, false><<<dim3(5, 5, NB), blk, 0, stream>>>(
        PHm, sPH, NCOL, H, sH, NP1, Abuf, sAA, DP1,
        nullptr, 0, 0, DP1, DP1, NCOL, 1.0f);

    // 3) C = A @ Q                      M=257 N=257 K=257
    wmma_gemm_f32<false, false><<<dim3(5, 5, NB), blk, 0, stream>>>(
        Abuf, sAA, DP1, Q, 0, DP1, Cbuf, sAA, DP1,
        nullptr, 0, 0, DP1, DP1, DP1, 1.0f);

    // 4) out = H + (C @ H) / n          M=257 N=2049 K=257, fused epilogue
    wmma_gemm_f32<false, true><<<dim3(5, 33, NB), blk, 0, stream>>>(
        Cbuf, sAA, DP1, H, sH, NP1, out, sH, NP1,
        H, sH, NP1, DP1, NP1, DP1, 1.0f / (float)NCOL);
}